// RGCNLayer_63617055588530
// MI455X (gfx1250) — compile-verified
//
#include <hip/hip_runtime.h>

typedef __attribute__((ext_vector_type(2))) float v2f;
typedef __attribute__((ext_vector_type(8))) float v8f;

#define ROWS_PER_WG    80
#define THREADS_PER_WG 160   // 5 waves of 32
#define KCHUNK         32
#define XPAD           36    // 32 + 4 -> conflict-free strided A-fragment b64 reads
#define WPITCH         288   // 2*128 + 32 floats per k-pair row: halves hit disjoint bank sets

// C[nrows x 128] = X[nrows x 128] * W[128 x 128] (+ bias), fp32 via V_WMMA_F32_16X16X4_F32.
// Each wave computes a 16-row x 128-col stripe as 8 accumulator tiles of 16x16.
// W is staged in LDS pair-interleaved along k so each B fragment is a single ds_load_b64.
__global__ __launch_bounds__(THREADS_PER_WG)
void rgcn_gemm_f32(const float* __restrict__ X, const float* __restrict__ W,
                   float* __restrict__ C, const float* __restrict__ bias)
{
    __shared__ float lx[ROWS_PER_WG * XPAD];          // 11.25 KB
    __shared__ float lw[(KCHUNK / 2) * WPITCH];       // 18 KB (16 k-pair rows)

    const int tid  = threadIdx.x;
    const int wave = tid >> 5;
    const int lane = tid & 31;
    const int half = lane >> 4;   // which 16-lane half
    const int m    = lane & 15;
    const int row0 = blockIdx.x * ROWS_PER_WG;
    const int wrow = wave * 16;   // this wave's row offset inside the WG tile

    v8f zero = {};
    v8f acc[8];
#pragma unroll
    for (int j = 0; j < 8; ++j) acc[j] = zero;

    for (int k0 = 0; k0 < 128; k0 += KCHUNK) {
        // Stage X chunk: ROWS_PER_WG x KCHUNK, row-major (pairs along k already adjacent)
        for (int idx = tid; idx < ROWS_PER_WG * KCHUNK; idx += THREADS_PER_WG) {
            int r = idx >> 5;            // / KCHUNK
            int c = idx & (KCHUNK - 1);
            lx[r * XPAD + c] = X[(size_t)(row0 + r) * 128 + k0 + c];
        }
        // Stage W chunk: KCHUNK x 128, pair-interleaved: (k,c) -> [(k>>1)*WPITCH + 2c + (k&1)]
        for (int idx = tid; idx < KCHUNK * 128; idx += THREADS_PER_WG) {
            int r = idx >> 7;
            int c = idx & 127;
            lw[(r >> 1) * WPITCH + (c << 1) + (r & 1)] = W[(size_t)(k0 + r) * 128 + c];
        }
        __syncthreads();

#pragma unroll
        for (int kk = 0; kk < KCHUNK; kk += 4) {
            // A fragment: 16x4 f32, lane(half,m): a[i] = X[wrow+m, kk + 2*half + i]
            // Adjacent floats, 8B aligned -> single ds_load_b64.
            v2f a = *(const v2f*)&lx[(wrow + m) * XPAD + kk + 2 * half];
#pragma unroll
            for (int j = 0; j < 8; ++j) {
                // B fragment: b[i] = W[kk + 2*half + i, 16j + m] -> one ds_load_b64
                v2f b = *(const v2f*)&lw[((kk >> 1) + half) * WPITCH + ((j * 16 + m) << 1)];
                acc[j] = __builtin_amdgcn_wmma_f32_16x16x4_f32(
                    false, a, false, b, (short)0, acc[j], false, false);
            }
        }
        __syncthreads();
    }

    // D layout: c[i] -> C[row0 + wrow + 8*half + i, 16j + m]
#pragma unroll
    for (int j = 0; j < 8; ++j) {
        int col = j * 16 + m;
        float bv = (bias != nullptr) ? bias[col] : 0.0f;
#pragma unroll
        for (int i = 0; i < 8; ++i) {
            int r = row0 + wrow + 8 * half + i;
            C[(size_t)r * 128 + col] = acc[j][i] + bv;
        }
    }
}

// One wave per edge; lanes cover 128 floats as float4; scatter-add into out[dst].
__global__ __launch_bounds__(256)
void rgcn_scatter(const float* __restrict__ hr, const int* __restrict__ edge_index,
                  const int* __restrict__ edge_type, float* __restrict__ out,
                  int num_edges, int rel)
{
    int e = blockIdx.x * 8 + (threadIdx.x >> 5);
    if (e >= num_edges) return;
    if (edge_type[e] != rel) return;

    int src = edge_index[e];              // row 0 of edge_index
    int dst = edge_index[num_edges + e];  // row 1
    int lane = threadIdx.x & 31;

    const float4 v = *(const float4*)(hr + (size_t)src * 128 + lane * 4);
    float* o = out + (size_t)dst * 128 + lane * 4;
    atomicAdd(o + 0, v.x);
    atomicAdd(o + 1, v.y);
    atomicAdd(o + 2, v.z);
    atomicAdd(o + 3, v.w);
}

__global__ __launch_bounds__(256)
void rgcn_relu(float* __restrict__ out, int n)
{
    int i = blockIdx.x * 256 + threadIdx.x;
    if (i < n) out[i] = fmaxf(out[i], 0.0f);
}

extern "C" void kernel_launch(void* const* d_in, const int* in_sizes, int n_in,
                              void* d_out, int out_size, void* d_ws, size_t ws_size,
                              hipStream_t stream) {
    const float* x     = (const float*)d_in[0];  // [50000,128]
    const float* w     = (const float*)d_in[1];  // [8,128,128]
    const float* selfw = (const float*)d_in[2];  // [128,128]
    const float* bias  = (const float*)d_in[3];  // [128]
    const int*   eidx  = (const int*)d_in[4];    // [2,E]
    const int*   etype = (const int*)d_in[5];    // [E]

    const int N = 50000;
    const int E = in_sizes[5];
    float* out = (float*)d_out;
    float* hr  = (float*)d_ws;                   // one relation's hr: N*128 f32 = 25.6 MB

    dim3 gemm_grid(N / ROWS_PER_WG);             // 625

    // out = x @ self_loop_weight + bias   (initializes the accumulator)
    rgcn_gemm_f32<<<gemm_grid, THREADS_PER_WG, 0, stream>>>(x, selfw, out, bias);

    // Relation-by-relation: hr_r = x @ W[r]; then scatter-add matching edges.
    int sc_blocks = (E + 7) / 8;
    for (int r = 0; r < 8; ++r) {
        rgcn_gemm_f32<<<gemm_grid, THREADS_PER_WG, 0, stream>>>(
            x, w + (size_t)r * 128 * 128, hr, nullptr);
        rgcn_scatter<<<sc_blocks, 256, 0, stream>>>(hr, eidx, etype, out, E, r);
    }

    int n = N * 128;
    rgcn_relu<<<(n + 255) / 256, 256, 0, stream>>>(out, n);
}